// MyModel_61933428416046
// MI455X (gfx1250) — compile-verified
//
#include <hip/hip_runtime.h>
#include <hip/hip_bf16.h>

typedef __attribute__((ext_vector_type(2))) float v2f;
typedef __attribute__((ext_vector_type(8))) float v8f;

#define MAX_SEQ_LEN 20
#define PAD_VALUE   60.0f
#define NUM_ROWS    1024
#define DIM         96
#define OUT_ELEMS   (NUM_ROWS * MAX_SEQ_LEN)   // 20480

// One wave32 computes 16 output elements. Gathered token vectors sit in the
// B matrix (one token per N column); A holds `ones` replicated across rows.
// D[m][n] = dot(inp[tok_n], ones) for every m, so lane l's c[0] is exactly
// the result for output slot (l & 15) -- no cross-lane traffic needed.
__global__ __launch_bounds__(256) void jagged_gather_dot_wmma(
    const float* __restrict__ inp,
    const float* __restrict__ ones,
    const int*   __restrict__ offsets,
    float*       __restrict__ out,
    int total_tokens)
{
    const int lane = threadIdx.x & 31;
    const int wave = (blockIdx.x * (blockDim.x >> 5)) + (threadIdx.x >> 5);
    const int base = wave * 16;            // first output element of this tile
    const int n    = lane & 15;            // this lane's output slot (N column)

    int e = base + n;                      // flat output element (grid sized exactly)
    const int b   = e / MAX_SEQ_LEN;
    const int j   = e - b * MAX_SEQ_LEN;
    const int s   = offsets[b];
    const int t   = offsets[b + 1];
    const int idx = s + j;
    const bool valid = (idx < t);
    int tok = valid ? idx : 0;
    if (tok > total_tokens - 1) tok = total_tokens - 1;   // mirrors jnp.clip

    // 32-bit A(16x4)/B(4x16) VGPR layout: lanes 0-15 carry K = k0,k0+1;
    // lanes 16-31 carry K = k0+2,k0+3  => contiguous float2 per lane.
    const int khalf = (lane >> 4) << 1;    // 0 or 2
    const float* rowp = inp  + (size_t)tok * DIM + khalf;  // B: token column
    const float* onep = ones + khalf;                      // A: ones rows

    v8f c = {};
#pragma unroll
    for (int k0 = 0; k0 < DIM; k0 += 4) {
        v2f a  = *(const v2f*)(onep + k0);
        v2f bm = *(const v2f*)(rowp + k0);
        // 8 args: (neg_a, A, neg_b, B, c_mod, C, reuse_a, reuse_b)
        c = __builtin_amdgcn_wmma_f32_16x16x4_f32(
                false, a, false, bm, (short)0, c, false, false);
    }

    // D layout: VGPR0, lane l (l<16) holds M=0, N=l  -> c[0] is our dot.
    if (lane < 16) {
        out[base + n] = valid ? c[0] : PAD_VALUE;
    }
}

extern "C" void kernel_launch(void* const* d_in, const int* in_sizes, int n_in,
                              void* d_out, int out_size, void* d_ws, size_t ws_size,
                              hipStream_t stream) {
    const float* inp     = (const float*)d_in[0];   // (1, T, 96) f32
    const float* ones    = (const float*)d_in[1];   // (1, 96, 1) f32
    const int*   offsets = (const int*)  d_in[2];   // (1025,) i32
    float*       out     = (float*)d_out;           // (1024, 20) f32

    const int total_tokens = in_sizes[0] / DIM;

    // 20480 outputs / 16 per wave = 1280 waves; 8 waves per 256-thread block.
    const int waves  = OUT_ELEMS / 16;
    const int blocks = waves / 8;          // 160
    jagged_gather_dot_wmma<<<blocks, 256, 0, stream>>>(
        inp, ones, offsets, out, total_tokens);
}